// CrossAttention1d_23321672417301
// MI455X (gfx1250) — compile-verified
//
#include <hip/hip_runtime.h>

// ---------------------------------------------------------------------------
// CrossAttention1d on MI455X (gfx1250, wave32, WMMA).
//
// Reassociation: out = (x^T cp) cr == x^T (cp cr) since there is no softmax.
// S = scale * cp_r @ cr is only [64x64] per batch -> ~48x less FLOPs and no
// 3072x3072 attention tensor. All GEMMs run on V_WMMA_F32_16X16X4_F32 to keep
// fp32 precision of the reference. Everything is L2-resident (~25MB total).
// ---------------------------------------------------------------------------

typedef __attribute__((ext_vector_type(2))) float v2f;
typedef __attribute__((ext_vector_type(8))) float v8f;

#define Bq   8
#define Cq   768
#define Nq   256
#define Hq   12
#define Dq   64
#define HN   3072     // H*N = H*Nc (flattened j / m axis)
#define BCN  196608   // per-batch element count: C*N = D*HN
#define KSL  16       // split-K slices for the S GEMM
#define KPS  192      // K per slice: 3072/16

static __device__ __forceinline__ v8f wmma4(v2f a, v2f b, v8f c) {
  // D = A(16x4,f32) * B(4x16,f32) + C(16x16,f32)
  return __builtin_amdgcn_wmma_f32_16x16x4_f32(false, a, false, b, (short)0, c,
                                               false, false);
}

// ---------------------------------------------------------------------------
// Kernel A: cp[b,o,n] = sum_c W_proj[o,c] * cross[b,c,n] + b_proj[o]
// One wave -> one 16x16 tile of cp[b]. K = 768.
// ---------------------------------------------------------------------------
__global__ void kA_proj(const float* __restrict__ Wp, const float* __restrict__ bp,
                        const float* __restrict__ cross, float* __restrict__ cp) {
  const int g    = blockIdx.x * (blockDim.x >> 5) + (threadIdx.x >> 5);
  const int lane = threadIdx.x & 31;
  const int half = lane >> 4;          // 0: K0/K1, 1: K2/K3 (ISA f32 A/B layout)
  const int l16  = lane & 15;
  const int koff = half << 1;

  const int b  = g / 768;              // 48*16 tiles per batch
  const int t  = g - b * 768;
  const int o0 = (t >> 4) << 4;
  const int n0 = (t & 15) << 4;

  const float* Arow  = Wp + (o0 + l16) * Cq;              // W_proj row, contiguous in k
  const float* Bcol  = cross + b * BCN + (n0 + l16);      // cross col, stride Nq in k

  v8f acc = {};
#pragma unroll 8
  for (int k = 0; k < Cq; k += 4) {
    v2f a = *(const v2f*)(Arow + k + koff);
    v2f bb;
    bb.x = Bcol[(k + koff) * Nq];
    bb.y = Bcol[(k + koff + 1) * Nq];
    acc = wmma4(a, bb, acc);
  }

  float* Cbase = cp + b * BCN + n0 + l16;
#pragma unroll
  for (int v = 0; v < 8; ++v) {
    const int row = o0 + v + (half << 3);
    Cbase[row * Nq] = acc[v] + bp[row];
  }
}

// ---------------------------------------------------------------------------
// Kernel B: split-K partials of S[b,d,e] = sum_j cp_r[b,d,j] * cr[b,j,e].
// cp_r[b] == cp[b] viewed as [64][3072]; cr[b,j,e] == cross[b*BCN + e*3072 + j]
// (the einops reshapes flatten to contiguous views). One wave -> one 16x16
// tile for one K-slice of 192. No atomics (deterministic).
// ---------------------------------------------------------------------------
__global__ void kB_spart(const float* __restrict__ cp, const float* __restrict__ cross,
                         float* __restrict__ spart) {
  const int g    = blockIdx.x * (blockDim.x >> 5) + (threadIdx.x >> 5);
  const int lane = threadIdx.x & 31;
  const int half = lane >> 4;
  const int l16  = lane & 15;
  const int koff = half << 1;

  const int b     = g >> 8;            // 16 tiles * 16 slices per batch
  const int r     = g & 255;
  const int tile  = r >> 4;
  const int slice = r & 15;
  const int d0    = (tile >> 2) << 4;
  const int e0    = (tile & 3) << 4;
  const int kb    = slice * KPS;

  const float* Arow = cp    + b * BCN + (d0 + l16) * HN + kb;  // contiguous in j
  const float* Brow = cross + b * BCN + (e0 + l16) * HN + kb;  // contiguous in j

  v8f acc = {};
#pragma unroll 8
  for (int k = 0; k < KPS; k += 4) {
    v2f a  = *(const v2f*)(Arow + k + koff);
    v2f bb = *(const v2f*)(Brow + k + koff);
    acc = wmma4(a, bb, acc);
  }

  float* Obase = spart + (((b << 4) + slice) << 12) + e0 + l16;
#pragma unroll
  for (int v = 0; v < 8; ++v) {
    Obase[(d0 + v + (half << 3)) << 6] = acc[v];
  }
}

// ---------------------------------------------------------------------------
// Deterministic split-K reduction + attention scale (D^-0.5 = 0.125).
// ---------------------------------------------------------------------------
__global__ void kRed(const float* __restrict__ spart, float* __restrict__ S) {
  const int i  = blockIdx.x * blockDim.x + threadIdx.x;   // 0..32767
  const int b  = i >> 12;
  const int de = i & 4095;
  float s = 0.0f;
#pragma unroll
  for (int sl = 0; sl < KSL; ++sl) s += spart[(((b << 4) + sl) << 12) + de];
  S[i] = s * 0.125f;
}

// ---------------------------------------------------------------------------
// Kernel C: OUT[b,m,dd] = sum_d XM[b,d,m] * S[b,d,dd],
// XM[b,d,m] == x_ori[b*BCN + d*3072 + m] (flattened view). K = 64.
// One wave -> one 16x16 tile of OUT[b] ([3072 x 64]).
// ---------------------------------------------------------------------------
__global__ void kC_out(const float* __restrict__ x, const float* __restrict__ S,
                       float* __restrict__ OUT) {
  const int g    = blockIdx.x * (blockDim.x >> 5) + (threadIdx.x >> 5);
  const int lane = threadIdx.x & 31;
  const int half = lane >> 4;
  const int l16  = lane & 15;
  const int koff = half << 1;

  const int b   = g / 768;             // 192 m-tiles * 4 dd-tiles per batch
  const int r   = g - b * 768;
  const int m0  = (r >> 2) << 4;
  const int dd0 = (r & 3) << 4;

  const float* Acol = x + b * BCN + (m0 + l16);        // stride HN in d
  const float* Bcol = S + (b << 12) + dd0 + l16;       // stride 64 in d

  v8f acc = {};
#pragma unroll
  for (int k = 0; k < Dq; k += 4) {
    v2f a, bb;
    a.x  = Acol[(k + koff) * HN];
    a.y  = Acol[(k + koff + 1) * HN];
    bb.x = Bcol[(k + koff) << 6];
    bb.y = Bcol[(k + koff + 1) << 6];
    acc = wmma4(a, bb, acc);
  }

  float* Obase = OUT + b * BCN + dd0 + l16;
#pragma unroll
  for (int v = 0; v < 8; ++v) {
    Obase[(m0 + v + (half << 3)) << 6] = acc[v];
  }
}

// ---------------------------------------------------------------------------
// Kernel D: final[b,c,n] = x_ori[b,c,n] + b_dep[c]
//                        + sum_{c'} W_dep[c,c'] * OUT_r[b,n,c'],
// where OUT_r[b] is OUT[b] viewed as [256][768] (the (N H) regroup flattens).
// K = 768. Fused bias + residual epilogue.
// ---------------------------------------------------------------------------
__global__ void kD_final(const float* __restrict__ Wd, const float* __restrict__ bd,
                         const float* __restrict__ OUT, const float* __restrict__ x,
                         float* __restrict__ y) {
  const int g    = blockIdx.x * (blockDim.x >> 5) + (threadIdx.x >> 5);
  const int lane = threadIdx.x & 31;
  const int half = lane >> 4;
  const int l16  = lane & 15;
  const int koff = half << 1;

  const int b  = g / 768;              // 48*16 tiles per batch
  const int t  = g - b * 768;
  const int c0 = (t >> 4) << 4;
  const int n0 = (t & 15) << 4;

  const float* Arow = Wd  + (c0 + l16) * Cq;           // W_dep row, contiguous
  const float* Brow = OUT + b * BCN + (n0 + l16) * Cq; // OUT_r row, contiguous

  v8f acc = {};
#pragma unroll 8
  for (int k = 0; k < Cq; k += 4) {
    v2f a  = *(const v2f*)(Arow + k + koff);
    v2f bb = *(const v2f*)(Brow + k + koff);
    acc = wmma4(a, bb, acc);
  }

#pragma unroll
  for (int v = 0; v < 8; ++v) {
    const int c   = c0 + v + (half << 3);
    const int idx = b * BCN + c * Nq + n0 + l16;
    y[idx] = x[idx] + bd[c] + acc[v];
  }
}

// ---------------------------------------------------------------------------
// Launch. Workspace layout (floats):
//   [0, 1572864)            cp   (reused as OUT after kernel B consumes it)
//   [1572864, 2097152)      spart: 8 batches x 16 slices x 64 x 64
//   [2097152, 2129920)      S:     8 x 64 x 64
// Total ~8.1 MB.
// ---------------------------------------------------------------------------
extern "C" void kernel_launch(void* const* d_in, const int* in_sizes, int n_in,
                              void* d_out, int out_size, void* d_ws, size_t ws_size,
                              hipStream_t stream) {
  (void)in_sizes; (void)n_in; (void)out_size; (void)ws_size;
  const float* x_ori = (const float*)d_in[0];
  const float* cross = (const float*)d_in[1];
  const float* Wp    = (const float*)d_in[2];
  const float* bp    = (const float*)d_in[3];
  const float* Wd    = (const float*)d_in[4];
  const float* bd    = (const float*)d_in[5];
  float* out = (float*)d_out;
  float* ws  = (float*)d_ws;

  float* cp    = ws;
  float* spart = ws + 1572864;
  float* S     = ws + 1572864 + 524288;
  float* OUT   = cp;  // cp is dead after kB; reuse its storage

  const dim3 blk(256);  // 8 wave32s per block
  kA_proj <<<768, blk, 0, stream>>>(Wp, bp, cross, cp);
  kB_spart<<<256, blk, 0, stream>>>(cp, cross, spart);
  kRed    <<<128, blk, 0, stream>>>(spart, S);
  kC_out  <<<768, blk, 0, stream>>>(x_ori, S, OUT);
  kD_final<<<768, blk, 0, stream>>>(Wd, bd, OUT, x_ori, out);
}